// PointCloudNormals_24455543783783
// MI455X (gfx1250) — compile-verified
//
#include <hip/hip_runtime.h>
#include <math.h>

typedef __attribute__((ext_vector_type(2))) float v2f;
typedef __attribute__((ext_vector_type(8))) float v8f;
typedef __attribute__((ext_vector_type(4))) int   v4i;
typedef __attribute__((address_space(1))) v4i     v4i_g;   // global int4
typedef __attribute__((address_space(3))) v4i     v4i_l;   // LDS int4

#define KNN   16
#define TILE  16
#define NMAX  8192
#define FBIG  3.402823466e+38f

// ---------------------------------------------------------------------------
// Kernel 0: pack (x,y,z) -> (x,y,z,|p|^2) float4 table in workspace.
// ---------------------------------------------------------------------------
__global__ void pack_kernel(const float* __restrict__ V,
                            float4* __restrict__ pts4, int total) {
  int t = blockIdx.x * blockDim.x + threadIdx.x;
  if (t >= total) return;
  float x = V[t * 3 + 0], y = V[t * 3 + 1], z = V[t * 3 + 2];
  pts4[t] = make_float4(x, y, z, x * x + y * y + z * z);
}

// ---------------------------------------------------------------------------
// Kernel 1: one wave32 per 16-query tile.
//   - whole batch point cloud async-copied into LDS (128KB of the 320KB WGP LDS)
//   - selection key s = |c|^2 - 2 c.q computed *inside* V_WMMA_F32_16X16X4_F32
//     by packing K=4 as A=(x,y,z,|c|^2), B=(-2qx,-2qy,-2qz,1)
//   - hot loop: 4 ds_load_b64 (1 wait) -> 4 back-to-back WMMAs -> 32 checks
//   - per-lane top-16 (LDS), merged across the lane pair holding each query
//   - weighted 3x3 covariance + closed-form smallest eigenvector -> normal
// ---------------------------------------------------------------------------
__global__ __launch_bounds__(32)
void knn_lrf_kernel(const float* __restrict__ pts4, float* __restrict__ normals,
                    int* __restrict__ knn, int N) {
  const int lane = threadIdx.x;    // 0..31
  const int hi   = lane >> 4;      // 0: lanes 0-15 (K=0,1), 1: lanes 16-31 (K=2,3)
  const int n16  = lane & 15;      // query column / candidate row this lane holds
  const int tilesPerBatch = N / TILE;
  const int b     = blockIdx.x / tilesPerBatch;
  const int qtile = blockIdx.x % tilesPerBatch;

  __shared__ float ldsP[NMAX * 4];       // packed (x,y,z,|p|^2) point cloud
  __shared__ float topD[32 * KNN];       // per-lane top-16 selection keys
  __shared__ int   topI[32 * KNN];       // per-lane top-16 indices
  __shared__ int   mrgI[TILE * KNN];     // merged final top-16 per query

  // ---- async preload of the entire batch into LDS (ASYNCcnt tracked) ----
  {
    const float* src = pts4 + (size_t)b * N * 4;
    for (int i = lane; i < N; i += 32) {
      __builtin_amdgcn_global_load_async_to_lds_b128(
          (v4i_g*)(src + (size_t)i * 4),
          (v4i_l*)(&ldsP[i * 4]), 0, 0);
    }
  }
#pragma unroll
  for (int k = 0; k < KNN; ++k) topD[lane * KNN + k] = FBIG;
  asm volatile("s_wait_asynccnt 0x0" ::: "memory");
  __syncthreads();

  // ---- B operand: queries (fixed for the wave) ----
  // lo lanes: (-2qx, -2qy)  hi lanes: (-2qz, 1)
  v2f qv = *(const v2f*)(ldsP + (size_t)(qtile * TILE + n16) * 4 + (hi ? 2 : 0));
  v2f bop;
  bop.x = -2.0f * qv.x;
  bop.y = hi ? 1.0f : (-2.0f * qv.y);

  float curMax = FBIG;
  int   curPos = 0;

  // per-result top-k update (rare insert path; rescan = 4x ds_load_b128)
  auto process = [&](const v8f& dmat, int tt) {
#pragma unroll
    for (int j = 0; j < 8; ++j) {
      const int m = j + (hi ? 8 : 0);            // candidate row in this tile
      float s = dmat[j];                         // |c|^2 - 2 c.q (ordering key)
      if (s < curMax) {
        topD[lane * KNN + curPos] = s;
        topI[lane * KNN + curPos] = tt * TILE + m;
        float mx = -FBIG; int mp = 0;
#pragma unroll
        for (int k = 0; k < KNN; ++k) {
          float dk = topD[lane * KNN + k];
          if (dk > mx) { mx = dk; mp = k; }
        }
        curMax = mx; curPos = mp;
      }
    }
  };

  // ---- main sweep: batch 4 LDS loads, then 4 WMMAs, then 32 checks ----
  const v8f zacc = {};
  int t = 0;
  for (; t + 4 <= tilesPerBatch; t += 4) {
    const float* base = ldsP + (size_t)(t * TILE + n16) * 4 + (hi ? 2 : 0);
    v2f a0 = *(const v2f*)(base + 0 * TILE * 4);
    v2f a1 = *(const v2f*)(base + 1 * TILE * 4);
    v2f a2 = *(const v2f*)(base + 2 * TILE * 4);
    v2f a3 = *(const v2f*)(base + 3 * TILE * 4);
    v8f d0 = __builtin_amdgcn_wmma_f32_16x16x4_f32(false, a0, false, bop,
                                                   (short)0, zacc, false, false);
    v8f d1 = __builtin_amdgcn_wmma_f32_16x16x4_f32(false, a1, false, bop,
                                                   (short)0, zacc, false, false);
    v8f d2 = __builtin_amdgcn_wmma_f32_16x16x4_f32(false, a2, false, bop,
                                                   (short)0, zacc, false, false);
    v8f d3 = __builtin_amdgcn_wmma_f32_16x16x4_f32(false, a3, false, bop,
                                                   (short)0, zacc, false, false);
    process(d0, t + 0);
    process(d1, t + 1);
    process(d2, t + 2);
    process(d3, t + 3);
  }
  for (; t < tilesPerBatch; ++t) {               // remainder (none for N=8192)
    v2f a0 = *(const v2f*)(ldsP + (size_t)(t * TILE + n16) * 4 + (hi ? 2 : 0));
    v8f d0 = __builtin_amdgcn_wmma_f32_16x16x4_f32(false, a0, false, bop,
                                                   (short)0, zacc, false, false);
    process(d0, t);
  }

  __syncthreads();

  // ---- merge lane pair lists, covariance, smallest eigenvector ----
  if (lane < TILE) {
#pragma unroll 1
    for (int k = 0; k < KNN; ++k) {
      float best = FBIG; int bs = lane * KNN;
#pragma unroll 1
      for (int s = 0; s < 32; ++s) {
        int row  = (s < 16) ? lane : (lane + 16);
        int slot = row * KNN + (s & 15);
        float dv = topD[slot];
        if (dv < best) { best = dv; bs = slot; }
      }
      mrgI[lane * KNN + k] = topI[bs];
      topD[bs] = FBIG;                   // exclude from further selection
    }

    const int   qg = qtile * TILE + lane;
    const float qx = ldsP[(size_t)qg * 4 + 0];
    const float qy = ldsP[(size_t)qg * 4 + 1];
    const float qz = ldsP[(size_t)qg * 4 + 2];

    float dist[KNN];
    float radii = 0.0f;
#pragma unroll
    for (int k = 0; k < KNN; ++k) {
      int id = mrgI[lane * KNN + k];
      float ex = ldsP[(size_t)id * 4 + 0] - qx;
      float ey = ldsP[(size_t)id * 4 + 1] - qy;
      float ez = ldsP[(size_t)id * 4 + 2] - qz;
      dist[k] = sqrtf(ex * ex + ey * ey + ez * ez);
      radii = fmaxf(radii, dist[k]);
    }
    float cxx = 0, cxy = 0, cxz = 0, cyy = 0, cyz = 0, czz = 0;
#pragma unroll
    for (int k = 0; k < KNN; ++k) {
      int id = mrgI[lane * KNN + k];
      float ex = ldsP[(size_t)id * 4 + 0] - qx;
      float ey = ldsP[(size_t)id * 4 + 1] - qy;
      float ez = ldsP[(size_t)id * 4 + 2] - qz;
      float w = radii - dist[k];
      cxx += w * ex * ex; cxy += w * ex * ey; cxz += w * ex * ez;
      cyy += w * ey * ey; cyz += w * ey * ez; czz += w * ez * ez;
    }

    // smallest eigenvector of symmetric 3x3 (sign-free: output uses n n^T)
    float nx = 1.0f, ny = 0.0f, nz = 0.0f;
    {
      float p1  = cxy * cxy + cxz * cxz + cyz * cyz;
      float qtr = (cxx + cyy + czz) * (1.0f / 3.0f);
      float b0 = cxx - qtr, b1 = cyy - qtr, b2 = czz - qtr;
      float p2 = b0 * b0 + b1 * b1 + b2 * b2 + 2.0f * p1;
      float pp = sqrtf(p2 * (1.0f / 6.0f));
      if (pp > 1e-30f) {
        float inv = 1.0f / pp;
        float c00 = b0 * inv, c01 = cxy * inv, c02 = cxz * inv;
        float c11 = b1 * inv, c12 = cyz * inv, c22 = b2 * inv;
        float detB = c00 * (c11 * c22 - c12 * c12)
                   - c01 * (c01 * c22 - c12 * c02)
                   + c02 * (c01 * c12 - c11 * c02);
        float r = fminf(1.0f, fmaxf(-1.0f, 0.5f * detB));
        float phi  = acosf(r) * (1.0f / 3.0f);
        float lmin = qtr + 2.0f * pp * cosf(phi + 2.0943951023931953f);
        float r0x = cxx - lmin, r0y = cxy,        r0z = cxz;
        float r1x = cxy,        r1y = cyy - lmin, r1z = cyz;
        float r2x = cxz,        r2y = cyz,        r2z = czz - lmin;
        float v0x = r0y * r1z - r0z * r1y, v0y = r0z * r1x - r0x * r1z, v0z = r0x * r1y - r0y * r1x;
        float v1x = r0y * r2z - r0z * r2y, v1y = r0z * r2x - r0x * r2z, v1z = r0x * r2y - r0y * r2x;
        float v2x = r1y * r2z - r1z * r2y, v2y = r1z * r2x - r1x * r2z, v2z = r1x * r2y - r1y * r2x;
        float q0 = v0x * v0x + v0y * v0y + v0z * v0z;
        float q1 = v1x * v1x + v1y * v1y + v1z * v1z;
        float q2 = v2x * v2x + v2y * v2y + v2z * v2z;
        float bx = v0x, by = v0y, bz = v0z, bn = q0;
        if (q1 > bn) { bx = v1x; by = v1y; bz = v1z; bn = q1; }
        if (q2 > bn) { bx = v2x; by = v2y; bz = v2z; bn = q2; }
        if (bn > 1e-30f) {
          float s = rsqrtf(bn);
          nx = bx * s; ny = by * s; nz = bz * s;
        }
      }
    }

    float* no = normals + (size_t)(b * N + qg) * 3;
    no[0] = nx; no[1] = ny; no[2] = nz;
    int* ko = knn + (size_t)(b * N + qg) * KNN;
#pragma unroll
    for (int k = 0; k < KNN; ++k) ko[k] = mrgI[lane * KNN + k];
  }
}

// ---------------------------------------------------------------------------
// Kernel 2: M = sum_k n_k n_k^T over neighbor normals, Frobenius-normalized.
// ---------------------------------------------------------------------------
__global__ void nncov_kernel(const float* __restrict__ normals,
                             const int* __restrict__ knn,
                             float* __restrict__ out, int N, int total) {
  int t = blockIdx.x * blockDim.x + threadIdx.x;
  if (t >= total) return;
  int b = t / N;
  const float* nb = normals + (size_t)b * N * 3;
  const int*   kk = knn + (size_t)t * KNN;
  float m00 = 0, m01 = 0, m02 = 0, m11 = 0, m12 = 0, m22 = 0;
#pragma unroll
  for (int k = 0; k < KNN; ++k) {
    int id = kk[k];
    const float* p = nb + (size_t)id * 3;
    float x = p[0], y = p[1], z = p[2];
    m00 += x * x; m01 += x * y; m02 += x * z;
    m11 += y * y; m12 += y * z; m22 += z * z;
  }
  float s = m00 * m00 + m11 * m11 + m22 * m22
          + 2.0f * (m01 * m01 + m02 * m02 + m12 * m12);
  float inv = rsqrtf(fmaxf(s, 1e-30f));
  float* o = out + (size_t)t * 9;
  o[0] = m00 * inv; o[1] = m01 * inv; o[2] = m02 * inv;
  o[3] = m01 * inv; o[4] = m11 * inv; o[5] = m12 * inv;
  o[6] = m02 * inv; o[7] = m12 * inv; o[8] = m22 * inv;
}

extern "C" void kernel_launch(void* const* d_in, const int* in_sizes, int n_in,
                              void* d_out, int out_size, void* d_ws, size_t ws_size,
                              hipStream_t stream) {
  const float* V = (const float*)d_in[0];
  const int B = 2;
  const int N = in_sizes[0] / (B * 3);        // 8192
  const int total = B * N;

  float* pts4    = (float*)d_ws;                                   // B*N*4 f32
  float* normals = pts4 + (size_t)total * 4;                       // B*N*3 f32
  int*   knn     = (int*)(normals + (size_t)total * 3);            // B*N*16 i32

  dim3 gp((total + 255) / 256), bp(256);
  hipLaunchKernelGGL(pack_kernel, gp, bp, 0, stream, V, (float4*)pts4, total);

  dim3 g1(B * (N / TILE)), b1(32);
  hipLaunchKernelGGL(knn_lrf_kernel, g1, b1, 0, stream, pts4, normals, knn, N);

  dim3 g2((total + 255) / 256), b2(256);
  hipLaunchKernelGGL(nncov_kernel, g2, b2, 0, stream, normals, knn,
                     (float*)d_out, N, total);
}